// AdaptiveMask_63617055589075
// MI455X (gfx1250) — compile-verified
//
#include <hip/hip_runtime.h>
#include <hip/hip_bf16.h>
#include <stdint.h>

// AdaptiveMask normalization: out = (x * mask) / (sum_lastaxis(x * mask) + 1e-8)
// x: (262144 rows, 289) f32.  mask built from current_val + 32-entry ramp template.
//
// Bandwidth-bound (606 MB traffic, ~0.3 GFLOP). Strategy: one wave per row,
// CDNA5 async global->LDS staging (ASYNCcnt) with double buffering, wave32
// shuffle reduction.

#define ROWLEN      289          // 17*17
#define SLICE_DW    320          // 10 * 32 dwords per row slice (rounded up)
#define WAVES_PB    8            // 256 threads / 32
#define LDS_DWORDS  (WAVES_PB * 2 * SLICE_DW)   // 5120 dw = 20 KB

__device__ __forceinline__ void stage_row_async(const float* __restrict__ xrow,
                                                uint32_t lds_base_bytes, int lane) {
#pragma unroll
    for (int it = 0; it < 10; ++it) {
        int p  = lane + (it << 5);
        int pc = (p > 288) ? 288 : p;       // clamp: duplicate tail dword, slot unused
        uint32_t ldsa = lds_base_bytes + ((uint32_t)p << 2);
        uint64_t ga   = (uint64_t)(uintptr_t)(xrow + pc);
        // gfx1250 async copy global -> LDS, tracked by ASYNCcnt
        asm volatile("global_load_async_to_lds_b32 %0, %1, off"
                     :: "v"(ldsa), "v"(ga) : "memory");
    }
}

__global__ __launch_bounds__(256) void
AdaptiveMask_63617055589075_kernel(const float* __restrict__ x,
                                   const float* __restrict__ current_val,
                                   const float* __restrict__ tmpl,
                                   float* __restrict__ out,
                                   long long rows, long long nwaves) {
    __shared__ float smem[LDS_DWORDS];

    const int lane = threadIdx.x & 31;
    const int w    = threadIdx.x >> 5;
    const long long wid = (long long)blockIdx.x * WAVES_PB + w;

    // ---- per-lane mask weights (row-invariant) -------------------------------
    // one_d[idx] = clip((tmpl[24+idx] + cv*32)/8 + 1, 0, 1); center (d==0) -> 1
    const float cv32 = current_val[0] * 32.0f;
    float m[10];
#pragma unroll
    for (int it = 0; it < 10; ++it) {
        int p = lane + (it << 5);
        if (p > 288) p = 288;
        int i  = p / 17;
        int j  = p - i * 17;
        int di = (i > 8) ? (i - 8) : (8 - i);
        int dj = (j > 8) ? (j - 8) : (8 - j);
        int d  = (di > dj) ? di : dj;       // Chebyshev distance, 0..8
        float mv;
        if (d == 0) {
            mv = 1.0f;
        } else {
            int idx = 8 - d;                // in [0,7]
            float t = tmpl[24 + idx];
            mv = (t + cv32) * 0.125f + 1.0f;
            mv = fminf(fmaxf(mv, 0.0f), 1.0f);
        }
        if (it == 9 && lane != 0) mv = 0.0f;  // only p==288 is real in last pass
        m[it] = mv;
    }

    // LDS byte bases of this wave's two staging buffers
    const uint32_t base0 = (uint32_t)(uintptr_t)(&smem[(w * 2 + 0) * SLICE_DW]);
    const uint32_t base1 = (uint32_t)(uintptr_t)(&smem[(w * 2 + 1) * SLICE_DW]);

    // ---- pipelined row loop --------------------------------------------------
    long long row = wid;
    if (row < rows) stage_row_async(x + row * ROWLEN, base0, lane);

    int buf = 0;
    for (; row < rows; row += nwaves) {
        const long long nxt = row + nwaves;
        const bool prefetch = (nxt < rows);
        if (prefetch)
            stage_row_async(x + nxt * ROWLEN, (buf ? base0 : base1), lane);

        // retire current buffer's 10 async loads (in-order retirement):
        if (prefetch) asm volatile("s_wait_asynccnt 0xa" ::: "memory");
        else          asm volatile("s_wait_asynccnt 0x0" ::: "memory");

        const float* slice = &smem[(w * 2 + buf) * SLICE_DW];
        float mv[10];
        float s = 0.0f;
#pragma unroll
        for (int it = 0; it < 10; ++it) {
            float v = slice[lane + (it << 5)];
            mv[it] = v * m[it];
            s += mv[it];
        }
        // wave32 butterfly reduction
#pragma unroll
        for (int off = 16; off; off >>= 1)
            s += __shfl_xor(s, off, 32);

        const float inv = 1.0f / (s + 1e-8f);

        float* __restrict__ orow = out + row * ROWLEN;
#pragma unroll
        for (int it = 0; it < 9; ++it)
            orow[lane + (it << 5)] = mv[it] * inv;   // p <= 287, coalesced
        if (lane == 0)
            orow[288] = mv[9] * inv;

        buf ^= 1;
    }
}

extern "C" void kernel_launch(void* const* d_in, const int* in_sizes, int n_in,
                              void* d_out, int out_size, void* d_ws, size_t ws_size,
                              hipStream_t stream) {
    const float* x    = (const float*)d_in[0];
    const float* cv   = (const float*)d_in[1];
    const float* tmpl = (const float*)d_in[2];
    float* out        = (float*)d_out;

    const long long rows = (long long)(in_sizes[0] / ROWLEN);   // 262144

    long long nblk = (rows + WAVES_PB - 1) / WAVES_PB;
    if (nblk > 2048) nblk = 2048;           // grid-stride; ~16 rows per wave
    const long long nwaves = nblk * WAVES_PB;

    AdaptiveMask_63617055589075_kernel<<<(int)nblk, 256, 0, stream>>>(
        x, cv, tmpl, out, rows, nwaves);
}